// SLabLinearAttention_7310034337878
// MI455X (gfx1250) — compile-verified
//
#include <hip/hip_runtime.h>

// SLab linear attention forward, fp32 end-to-end.
// Matrix ops: CDNA5 fp32 WMMA pipe (v_wmma_f32_16x16x4_f32).
// Data movement: GLOBAL_LOAD_ASYNC_TO_LDS_B128 + ASYNCcnt double buffering.
// B=4, N=4096, C=1024, H=16, HD=64.

typedef __attribute__((ext_vector_type(2))) float v2f;
typedef __attribute__((ext_vector_type(8))) float v8f;
typedef int v4i __attribute__((vector_size(16)));

#define DEVINL __device__ __forceinline__

DEVINL v8f wmma4(v2f a, v2f b, v8f c) {
  // D = A(16x4 f32) x B(4x16 f32) + C(16x16 f32)
  return __builtin_amdgcn_wmma_f32_16x16x4_f32(
      /*neg_a=*/false, a, /*neg_b=*/false, b,
      /*c_mod=*/(short)0, c, /*reuse_a=*/false, /*reuse_b=*/false);
}

DEVINL v8f v8f_zero() { v8f z = {0.f,0.f,0.f,0.f,0.f,0.f,0.f,0.f}; return z; }

// ---- CDNA5 async global->LDS copy (16B per lane), ASYNCcnt-tracked --------
typedef __attribute__((address_space(1))) v4i* gvec_p;  // global v4i*
typedef __attribute__((address_space(3))) v4i* lvec_p;  // LDS v4i*

#if __has_builtin(__builtin_amdgcn_global_load_async_to_lds_b128)
DEVINL void async_copy_b128(const float* gp, float* lp) {
  __builtin_amdgcn_global_load_async_to_lds_b128(
      (gvec_p)(unsigned long long)gp,
      (lvec_p)(unsigned long long)lp,
      /*imm offset=*/0, /*cpol=*/0);
}
#else
DEVINL void async_copy_b128(const float* gp, float* lp) {
  // flat LDS address low 32 bits == LDS byte offset (aperture truncation)
  asm volatile("global_load_async_to_lds_b128 %0, %1, off"
               :: "v"((unsigned)(unsigned long long)lp), "v"(gp)
               : "memory");
}
#endif

#if __has_builtin(__builtin_amdgcn_s_wait_asynccnt)
#define WAIT_ASYNC(n) __builtin_amdgcn_s_wait_asynccnt(n)
#else
#define WAIT_ASYNC(n) asm volatile("s_wait_asynccnt %0" :: "i"(n) : "memory")
#endif

// ---------------------------------------------------------------------------
// Kernel 1: QKV GEMM. C[m,col] = sum_k X[m,k]*W[col,k] + b[col], then
//   q = relu, k = relu(+pos_enc), v = raw, scattered into head layout
//   (bh=64, n=4096, hd=64).
// Tile: 64(M) x 128(N) per block, KC=32, double-buffered async staging.
// ---------------------------------------------------------------------------
__global__ __launch_bounds__(256)
void qkv_gemm_kernel(const float* __restrict__ X, const float* __restrict__ W,
                     const float* __restrict__ bias, const float* __restrict__ pos_enc,
                     float* __restrict__ qh, float* __restrict__ kh,
                     float* __restrict__ vh) {
  const int K = 1024;
  const int LP = 36;  // padded LDS row stride (conflict-free frag reads)
  __shared__ float lx[2][64 * 36];
  __shared__ float lw[2][128 * 36];

  const int tid  = threadIdx.x;
  const int lane = tid & 31;
  const int wave = tid >> 5;
  const int n0 = blockIdx.x * 128;
  const int m0 = blockIdx.y * 64;
  const int ms  = (wave & 3) * 16;   // M strip within tile
  const int nsb = (wave >> 2) * 64;  // N half within tile

  v8f acc[4];
  #pragma unroll
  for (int j = 0; j < 4; ++j) acc[j] = v8f_zero();

  // 6 async B128 per thread per chunk (2 for X 64x32, 4 for W 128x32)
  auto stage = [&](int k0, int buf) {
    #pragma unroll
    for (int t = 0; t < 2; ++t) {
      const int idx = tid + t * 256;
      const int row = idx >> 3, c4 = (idx & 7) * 4;
      async_copy_b128(&X[(size_t)(m0 + row) * K + k0 + c4],
                      &lx[buf][row * LP + c4]);
    }
    #pragma unroll
    for (int t = 0; t < 4; ++t) {
      const int idx = tid + t * 256;
      const int row = idx >> 3, c4 = (idx & 7) * 4;
      async_copy_b128(&W[(size_t)(n0 + row) * K + k0 + c4],
                      &lw[buf][row * LP + c4]);
    }
  };

  const int NCH = K / 32;  // 32 chunks
  stage(0, 0);
  for (int i = 0; i < NCH; ++i) {
    const int cur = i & 1;
    if (i + 1 < NCH) {
      stage((i + 1) * 32, cur ^ 1);
      WAIT_ASYNC(6);   // chunk i complete; chunk i+1 still in flight
    } else {
      WAIT_ASYNC(0);
    }
    __syncthreads();

    const int khalf = (lane >> 4) * 2;
    const int arow  = (ms + (lane & 15)) * LP;
    #pragma unroll
    for (int kk = 0; kk < 32; kk += 4) {
      v2f a = *(const v2f*)&lx[cur][arow + kk + khalf];
      #pragma unroll
      for (int j = 0; j < 4; ++j) {
        v2f b = *(const v2f*)&lw[cur][(nsb + j * 16 + (lane & 15)) * LP + kk + khalf];
        acc[j] = wmma4(a, b, acc[j]);
      }
    }
    __syncthreads();
  }

  // Epilogue: bias (+pos_enc), relu, scatter to head layout.
  const int mrow = m0 + ms + ((lane >> 4) << 3);  // +8 for upper half-wave
  #pragma unroll
  for (int j = 0; j < 4; ++j) {
    const int col   = n0 + nsb + j * 16 + (lane & 15);
    const int which = col >> 10;      // 0=q 1=k 2=v (uniform per subtile)
    const int cc    = col & 1023;
    const int h  = cc >> 6;
    const int hd = cc & 63;
    const float bval = bias[col];
    const float pe   = (which == 1) ? pos_enc[cc] : 0.f;
    #pragma unroll
    for (int r = 0; r < 8; ++r) {
      const int m    = mrow + r;
      const int bidx = m >> 12;
      const int nn   = m & 4095;
      const size_t addr = ((size_t)(bidx * 16 + h) * 4096 + nn) * 64 + hd;
      const float val = acc[j][r] + bval;
      if (which == 0)      qh[addr] = fmaxf(val, 0.f);
      else if (which == 1) kh[addr] = fmaxf(val + pe, 0.f);
      else                 vh[addr] = val;
    }
  }
}

// ---------------------------------------------------------------------------
// Kernel 2: ksum[head][hd] = sum_n k[head][n][hd]
// ---------------------------------------------------------------------------
__global__ __launch_bounds__(256)
void ksum_kernel(const float* __restrict__ kh, float* __restrict__ ksum) {
  __shared__ float red[256];
  const int head = blockIdx.x;
  const int tid = threadIdx.x;
  const int hd = tid & 63;
  const int seg = tid >> 6;
  float s = 0.f;
  const float* base = kh + (size_t)head * 4096 * 64 + hd;
  for (int n = seg; n < 4096; n += 4) s += base[(size_t)n * 64];
  red[tid] = s;
  __syncthreads();
  if (seg == 0)
    ksum[head * 64 + hd] = red[hd] + red[hd + 64] + red[hd + 128] + red[hd + 192];
}

// ---------------------------------------------------------------------------
// Kernel 3: z[head][n] = 1 / (q[head][n] . ksum[head] + 1e-6)
// ---------------------------------------------------------------------------
__global__ __launch_bounds__(256)
void z_kernel(const float* __restrict__ qh, const float* __restrict__ ksum,
              float* __restrict__ zbuf) {
  __shared__ float ks[64];
  const int head = blockIdx.y;
  const int tid = threadIdx.x;
  if (tid < 64) ks[tid] = ksum[head * 64 + tid];
  __syncthreads();
  const int n = blockIdx.x * 256 + tid;
  const float4* qr = (const float4*)(qh + ((size_t)head * 4096 + n) * 64);
  float s = 0.f;
  #pragma unroll
  for (int c4 = 0; c4 < 16; ++c4) {
    const float4 qv = qr[c4];
    s += qv.x * ks[c4 * 4 + 0] + qv.y * ks[c4 * 4 + 1] +
         qv.z * ks[c4 * 4 + 2] + qv.w * ks[c4 * 4 + 3];
  }
  zbuf[(size_t)head * 4096 + n] = 1.f / (s + 1e-6f);
}

// ---------------------------------------------------------------------------
// Kernel 4: kv[head] = k[head]^T @ v[head] (64x64), one block per head.
// NC=32 n-chunks, double-buffered async staging; A frag read transposed.
// ---------------------------------------------------------------------------
__global__ __launch_bounds__(256)
void kv_kernel(const float* __restrict__ kh, const float* __restrict__ vh,
               float* __restrict__ kvbuf) {
  __shared__ float lk[2][32 * 68];
  __shared__ float lv[2][32 * 68];
  const int head = blockIdx.x;
  const int tid = threadIdx.x;
  const int lane = tid & 31;
  const int wave = tid >> 5;
  const int cs  = (wave & 3) * 16;   // output row (c) strip
  const int ds0 = (wave >> 2) * 32;  // output col (d) half
  v8f acc[2] = {v8f_zero(), v8f_zero()};
  const size_t hbase = (size_t)head * 4096 * 64;

  // 4 async B128 per thread per chunk (2 for k 32x64, 2 for v 32x64)
  auto stage = [&](int n0, int buf) {
    #pragma unroll
    for (int t = 0; t < 2; ++t) {
      const int idx = tid + t * 256;
      const int row = idx >> 4, c4 = (idx & 15) * 4;
      async_copy_b128(&kh[hbase + (size_t)(n0 + row) * 64 + c4],
                      &lk[buf][row * 68 + c4]);
      async_copy_b128(&vh[hbase + (size_t)(n0 + row) * 64 + c4],
                      &lv[buf][row * 68 + c4]);
    }
  };

  stage(0, 0);
  for (int i = 0; i < 128; ++i) {
    const int cur = i & 1;
    if (i + 1 < 128) {
      stage((i + 1) * 32, cur ^ 1);
      WAIT_ASYNC(4);
    } else {
      WAIT_ASYNC(0);
    }
    __syncthreads();

    const int khalf = (lane >> 4) * 2;
    const int c = cs + (lane & 15);
    #pragma unroll
    for (int kk = 0; kk < 32; kk += 4) {
      const int r0 = kk + khalf;
      v2f a; a.x = lk[cur][r0 * 68 + c]; a.y = lk[cur][(r0 + 1) * 68 + c];
      #pragma unroll
      for (int j = 0; j < 2; ++j) {
        const int d = ds0 + j * 16 + (lane & 15);
        v2f b; b.x = lv[cur][r0 * 68 + d]; b.y = lv[cur][(r0 + 1) * 68 + d];
        acc[j] = wmma4(a, b, acc[j]);
      }
    }
    __syncthreads();
  }

  const int crow = cs + ((lane >> 4) << 3);
  #pragma unroll
  for (int j = 0; j < 2; ++j) {
    const int d = ds0 + j * 16 + (lane & 15);
    #pragma unroll
    for (int r = 0; r < 8; ++r)
      kvbuf[(size_t)head * 4096 + (size_t)(crow + r) * 64 + d] = acc[j][r];
  }
}

// ---------------------------------------------------------------------------
// Kernel 5: o1[head][n][hd] = (q[head] @ kv[head])[n][hd] * z[head][n]
// o1 aliases qh: q tile fully staged (wait+barrier) before any write.
// ---------------------------------------------------------------------------
__global__ __launch_bounds__(256)
void attn_kernel(const float* qh, const float* __restrict__ kvbuf,
                 const float* __restrict__ zbuf, float* o1) {
  __shared__ float lq[64 * 68];
  __shared__ float lkv[64 * 68];
  const int head = blockIdx.y;
  const int m0 = blockIdx.x * 64;
  const int tid = threadIdx.x;
  const int lane = tid & 31;
  const int wave = tid >> 5;
  const int srow = tid >> 4;
  const int scol = (tid & 15) * 4;
  const size_t qbase = ((size_t)head * 4096 + m0) * 64;

  #pragma unroll
  for (int r = 0; r < 4; ++r) {
    const int row = srow + r * 16;
    async_copy_b128(&qh[qbase + (size_t)row * 64 + scol], &lq[row * 68 + scol]);
    async_copy_b128(&kvbuf[(size_t)head * 4096 + (size_t)row * 64 + scol],
                    &lkv[row * 68 + scol]);
  }
  WAIT_ASYNC(0);
  __syncthreads();

  const int ms  = (wave & 3) * 16;
  const int ns0 = (wave >> 2) * 32;
  v8f acc[2] = {v8f_zero(), v8f_zero()};
  const int khalf = (lane >> 4) * 2;
  const int arow = (ms + (lane & 15)) * 68;
  #pragma unroll 4
  for (int kk = 0; kk < 64; kk += 4) {
    v2f a = *(const v2f*)&lq[arow + kk + khalf];
    #pragma unroll
    for (int j = 0; j < 2; ++j) {
      const int d = ns0 + j * 16 + (lane & 15);
      v2f b; b.x = lkv[(kk + khalf) * 68 + d]; b.y = lkv[(kk + khalf + 1) * 68 + d];
      acc[j] = wmma4(a, b, acc[j]);
    }
  }

  const int mrow = ms + ((lane >> 4) << 3);
  #pragma unroll
  for (int r = 0; r < 8; ++r) {
    const int m = m0 + mrow + r;
    const float zz = zbuf[(size_t)head * 4096 + m];
    #pragma unroll
    for (int j = 0; j < 2; ++j) {
      const int d = ns0 + j * 16 + (lane & 15);
      o1[((size_t)head * 4096 + m) * 64 + d] = acc[j][r] * zz;
    }
  }
}

// ---------------------------------------------------------------------------
// Kernel 6: depthwise conv: width dim is 1 so only kernel column kw=2 lands
// in-bounds -> 5-tap conv along n with taps dwc_w[hd][0][t][2]; add in place.
// ---------------------------------------------------------------------------
__global__ __launch_bounds__(256)
void dwconv_add_kernel(const float* __restrict__ vh, const float* __restrict__ dwc_w,
                       const float* __restrict__ dwc_b, float* __restrict__ o1) {
  const size_t idx = (size_t)blockIdx.x * 256 + threadIdx.x;
  const int hd = (int)(idx & 63);
  const int n = (int)((idx >> 6) & 4095);
  const int head = (int)(idx >> 18);
  float s = dwc_b[hd];
  const float* vcol = vh + (size_t)head * 4096 * 64 + hd;
  #pragma unroll
  for (int t = 0; t < 5; ++t) {
    const int nn = n + t - 2;
    if (nn >= 0 && nn < 4096)
      s += dwc_w[hd * 25 + t * 5 + 2] * vcol[(size_t)nn * 64];
  }
  o1[idx] += s;
}

// ---------------------------------------------------------------------------
// Kernel 7: proj GEMM. Input is o1 in head layout; a 32-wide k-chunk is a
// contiguous 64x32 sub-block of one head. out = o1_view @ proj_w^T + b.
// ---------------------------------------------------------------------------
__global__ __launch_bounds__(256)
void proj_gemm_kernel(const float* __restrict__ o1, const float* __restrict__ W,
                      const float* __restrict__ bias, float* __restrict__ out) {
  const int K = 1024;
  const int LP = 36;
  __shared__ float lx[2][64 * 36];
  __shared__ float lw[2][128 * 36];
  const int tid = threadIdx.x;
  const int lane = tid & 31;
  const int wave = tid >> 5;
  const int n0 = blockIdx.x * 128;
  const int m0 = blockIdx.y * 64;
  const int ms  = (wave & 3) * 16;
  const int nsb = (wave >> 2) * 64;
  v8f acc[4];
  #pragma unroll
  for (int j = 0; j < 4; ++j) acc[j] = v8f_zero();
  const int bb = m0 >> 12;
  const int nnb = m0 & 4095;

  auto stage = [&](int k0, int buf) {
    const int h = k0 >> 6;
    const float* xs = o1 + ((size_t)(bb * 16 + h) * 4096 + nnb) * 64 + (k0 & 63);
    #pragma unroll
    for (int t = 0; t < 2; ++t) {
      const int idx = tid + t * 256;
      const int row = idx >> 3, c4 = (idx & 7) * 4;
      async_copy_b128(&xs[(size_t)row * 64 + c4], &lx[buf][row * LP + c4]);
    }
    #pragma unroll
    for (int t = 0; t < 4; ++t) {
      const int idx = tid + t * 256;
      const int row = idx >> 3, c4 = (idx & 7) * 4;
      async_copy_b128(&W[(size_t)(n0 + row) * K + k0 + c4],
                      &lw[buf][row * LP + c4]);
    }
  };

  const int NCH = K / 32;
  stage(0, 0);
  for (int i = 0; i < NCH; ++i) {
    const int cur = i & 1;
    if (i + 1 < NCH) {
      stage((i + 1) * 32, cur ^ 1);
      WAIT_ASYNC(6);
    } else {
      WAIT_ASYNC(0);
    }
    __syncthreads();

    const int khalf = (lane >> 4) * 2;
    const int arow = (ms + (lane & 15)) * LP;
    #pragma unroll
    for (int kk = 0; kk < 32; kk += 4) {
      v2f a = *(const v2f*)&lx[cur][arow + kk + khalf];
      #pragma unroll
      for (int j = 0; j < 4; ++j) {
        v2f b = *(const v2f*)&lw[cur][(nsb + j * 16 + (lane & 15)) * LP + kk + khalf];
        acc[j] = wmma4(a, b, acc[j]);
      }
    }
    __syncthreads();
  }

  const int mrow = m0 + ms + ((lane >> 4) << 3);
  #pragma unroll
  for (int j = 0; j < 4; ++j) {
    const int col = n0 + nsb + j * 16 + (lane & 15);
    const float bval = bias[col];
    #pragma unroll
    for (int r = 0; r < 8; ++r)
      out[(size_t)(mrow + r) * 1024 + col] = acc[j][r] + bval;
  }
}

// ---------------------------------------------------------------------------
extern "C" void kernel_launch(void* const* d_in, const int* in_sizes, int n_in,
                              void* d_out, int out_size, void* d_ws, size_t ws_size,
                              hipStream_t stream) {
  (void)in_sizes; (void)n_in; (void)out_size; (void)ws_size;
  const float* x      = (const float*)d_in[0];
  const float* qkv_w  = (const float*)d_in[1];
  const float* qkv_b  = (const float*)d_in[2];
  const float* proj_w = (const float*)d_in[3];
  const float* proj_b = (const float*)d_in[4];
  const float* pos    = (const float*)d_in[5];
  const float* dwc_w  = (const float*)d_in[6];
  const float* dwc_b  = (const float*)d_in[7];
  float* out = (float*)d_out;

  float* ws = (float*)d_ws;
  const size_t HNHD = 64ull * 4096 * 64;  // per-tensor head-layout size
  float* qh   = ws;                        // also reused as o1 after q consumed
  float* kh   = qh + HNHD;
  float* vh   = kh + HNHD;
  float* kv   = vh + HNHD;                 // 64 * 64 * 64
  float* ksum = kv + 64 * 64 * 64;         // 64 * 64
  float* zb   = ksum + 64 * 64;            // 64 * 4096
  float* o1   = qh;                        // alias: q fully consumed before writes

  qkv_gemm_kernel<<<dim3(24, 256), dim3(256), 0, stream>>>(x, qkv_w, qkv_b, pos,
                                                           qh, kh, vh);
  ksum_kernel<<<dim3(64), dim3(256), 0, stream>>>(kh, ksum);
  z_kernel<<<dim3(16, 64), dim3(256), 0, stream>>>(qh, ksum, zb);
  kv_kernel<<<dim3(64), dim3(256), 0, stream>>>(kh, vh, kv);
  attn_kernel<<<dim3(64, 64), dim3(256), 0, stream>>>(qh, kv, zb, o1);
  dwconv_add_kernel<<<dim3(65536), dim3(256), 0, stream>>>(vh, dwc_w, dwc_b, o1);
  proj_gemm_kernel<<<dim3(8, 256), dim3(256), 0, stream>>>(o1, proj_w, proj_b, out);
}